// WaveNet_10934986735921
// MI455X (gfx1250) — compile-verified
//
#include <hip/hip_runtime.h>

// CDNA5 / gfx1250, wave32. fp32 WMMA 16x16x4 used as im2col GEMM for the
// 4-channel 4x4 stride-2 db2 analysis filter bank. Memory-bound problem
// (~0.4 GB traffic vs ~6 GFLOP), so fp32 WMMA + fused quadrant scatter.

typedef float v2f __attribute__((ext_vector_type(2)));
typedef float v8f __attribute__((ext_vector_type(8)));

// One wave handles a 16-wide column tile for 8 consecutive output rows of one
// batch image at one level. Block = 256 threads = 8 independent waves.
__global__ __launch_bounds__(256) void dwt_db2_level_kernel(
    const float* __restrict__ src,   // (B, 2S, 2S) input of this level
    float* __restrict__ out,         // (B, 2048, 2048) final mosaic
    float* __restrict__ llout,       // (B, S, S) LL scratch; nullptr -> LL to mosaic (last level)
    const float* __restrict__ Wg,    // (4,1,4,4) OIHW filters, c*16 + p*4 + q
    const float* __restrict__ bias,  // (4,)
    int S,                           // quadrant (output) size = Hin/2
    int tiles_x,                     // S / 16
    int tiles_y)                     // S / 8
{
    const int lane = threadIdx.x & 31;
    const int wid  = blockIdx.x * 8 + (threadIdx.x >> 5);

    const int jt  = wid % tiles_x;
    int tmp       = wid / tiles_x;
    const int iy  = tmp % tiles_y;
    const int b   = tmp / tiles_y;

    const int Hin = 2 * S, Win = 2 * S;
    const int n   = lane & 15;   // N index (output column within tile) / A row m
    const int h   = lane >> 4;   // lane half selects K pair {2h, 2h+1}

    // ---- A matrix: Wmat[m][k] for chunk kc = filter row p = kc, tap q = k.
    // Layout (ISA 7.12.2, 32-bit A 16x4): lanes 0-15 -> VGPR0=K0, VGPR1=K1;
    // lanes 16-31 -> VGPR0=K2, VGPR1=K3. Rows m>=4 are zero.
    const int  m    = n;
    const bool mok  = (m < 4);
    const int  msel = mok ? m : 0;
    v2f a[4];
#pragma unroll
    for (int kc = 0; kc < 4; ++kc) {
        float x0 = Wg[msel * 16 + kc * 4 + 2 * h];
        float x1 = Wg[msel * 16 + kc * 4 + 2 * h + 1];
        a[kc].x = mok ? x0 : 0.0f;
        a[kc].y = mok ? x1 : 0.0f;
    }
    const float bz0 = bias[0], bz1 = bias[1], bz2 = bias[2], bz3 = bias[3];

    const float* __restrict__ srcb = src + (long)b * Hin * Win;
    const int j0 = jt * 16;
    const int i0 = iy * 8;

    // B-matrix column for this lane: out col j0+n, tap cols 2(j0+n)-1+{2h,2h+1}
    const int c0 = 2 * (j0 + n) - 1 + 2 * h;
    const int c1 = c0 + 1;
    const bool c0in = (unsigned)c0 < (unsigned)Win;
    const bool c1in = (unsigned)c1 < (unsigned)Win;

    const long obase = (long)b * (2048 * 2048);

    for (int di = 0; di < 8; ++di) {
        const int i = i0 + di;
        v8f acc = {};
#pragma unroll
        for (int kc = 0; kc < 4; ++kc) {
            // input row for filter row p = kc (zero-pad 1): r = 2i + kc - 1
            const int  r   = 2 * i + kc - 1;
            const bool rok = (unsigned)r < (unsigned)Hin;
            const bool ok0 = rok & c0in;
            const bool ok1 = rok & c1in;
            // clamp addresses (keep EXEC all-ones for WMMA; zero via select)
            const long base = (long)(rok ? r : 0) * Win;
            float v0 = srcb[ok0 ? (base + c0) : 0];
            float v1 = srcb[ok1 ? (base + c1) : 0];
            v0 = ok0 ? v0 : 0.0f;
            v1 = ok1 ? v1 : 0.0f;
            v2f bb; bb.x = v0; bb.y = v1;
            // D(4x16 used) += Wchunk(4x4) * Patch(4x16)
            acc = __builtin_amdgcn_wmma_f32_16x16x4_f32(
                false, a[kc], false, bb, (short)0, acc, false, false);
        }
        // D layout: acc[r] = channel r, column j0+lane, for lanes 0-15.
        if (lane < 16) {
            const int  j  = j0 + n;
            const float ll = acc[0] + bz0;
            const float lh = acc[1] + bz1;
            const float hl = acc[2] + bz2;
            const float hh = acc[3] + bz3;
            out[obase + (long)i * 2048 + (j + S)]       = lh;  // top-right
            out[obase + (long)(i + S) * 2048 + j]       = hl;  // bottom-left
            out[obase + (long)(i + S) * 2048 + (j + S)] = hh;  // bottom-right
            if (llout) llout[(long)b * S * S + (long)i * S + j] = ll;
            else       out[obase + (long)i * 2048 + j]          = ll; // last level
        }
    }
}

extern "C" void kernel_launch(void* const* d_in, const int* in_sizes, int n_in,
                              void* d_out, int out_size, void* d_ws, size_t ws_size,
                              hipStream_t stream) {
    (void)in_sizes; (void)n_in; (void)out_size; (void)ws_size;
    const float* vimg = (const float*)d_in[0];
    // d_in[1] = levels (==3, fixed by setup; device scalar, hardcoded here)
    const float* Wg   = (const float*)d_in[2];
    const float* bias = (const float*)d_in[3];
    float* out = (float*)d_out;

    const int B = 8;
    float* ll1 = (float*)d_ws;                       // 8*1024*1024 f32 = 32 MB
    float* ll2 = ll1 + (size_t)B * 1024 * 1024;      // 8*512*512   f32 =  8 MB

    // Level 1: 2048 -> 1024 quadrants; LL -> ll1
    {
        const int S = 1024, tx = S / 16, ty = S / 8;
        const int waves = B * tx * ty;
        dwt_db2_level_kernel<<<waves / 8, 256, 0, stream>>>(
            vimg, out, ll1, Wg, bias, S, tx, ty);
    }
    // Level 2: 1024 -> 512 quadrants; LL -> ll2
    {
        const int S = 512, tx = S / 16, ty = S / 8;
        const int waves = B * tx * ty;
        dwt_db2_level_kernel<<<waves / 8, 256, 0, stream>>>(
            ll1, out, ll2, Wg, bias, S, tx, ty);
    }
    // Level 3: 512 -> 256 quadrants; LL goes to the mosaic (top-left corner)
    {
        const int S = 256, tx = S / 16, ty = S / 8;
        const int waves = B * tx * ty;
        dwt_db2_level_kernel<<<waves / 8, 256, 0, stream>>>(
            ll2, out, nullptr, Wg, bias, S, tx, ty);
    }
}